// CentroidEstimator_40355512713832
// MI455X (gfx1250) — compile-verified
//
#include <hip/hip_runtime.h>

typedef __attribute__((ext_vector_type(2))) float v2f;
typedef __attribute__((ext_vector_type(8))) float v8f;

constexpr float kAlpha = 0.9f;
constexpr float kEps   = 1e-3f;
constexpr int kB = 4096;
constexpr int kF = 512;
constexpr int kK = 64;
constexpr int kD = 4;
constexpr int kBSplit = 8;               // split-B factor
constexpr int kBChunk = kB / kBSplit;    // 512 rows of B per wave
constexpr int kFTiles = kF / 16;         // 32
constexpr int kKTiles = kK / 16;         // 4

// ---------------------------------------------------------------------------
// Kernel 1: domain-masked split-B GEMM accumulation via V_WMMA_F32_16X16X4_F32.
// One wave32 per (f-tile, k-tile, b-split). Accumulates
//   num_d[d, f0+m, k0+n] += sum_{b in split, dom[b]==d} features[b,f0+m]*p[b,k0+n]
//   den_d[d, k0+n]       += sum_{b in split, dom[b]==d} p[b,k0+n]   (f-tile 0 only)
// into workspace with f32 global atomics.
//
// f32 WMMA fragment layouts (ISA 7.12.2):
//   A 16x4 (MxK): lanes 0-15 -> M=lane, v0:K=0 v1:K=1; lanes 16-31 -> v0:K=2 v1:K=3
//   B 4x16 (KxN): lanes 0-15 -> N=lane, v0:K=0 v1:K=1; lanes 16-31 -> v0:K=2 v1:K=3
//   C/D 16x16:    VGPR r, lanes 0-15 -> (M=r,   N=lane)
//                          lanes 16-31 -> (M=r+8, N=lane-16)
// ---------------------------------------------------------------------------
__global__ __launch_bounds__(32)
void centroid_accum_wmma(const float* __restrict__ features,   // (B, F)
                         const int*   __restrict__ domains,    // (B,)
                         const float* __restrict__ probs,      // (B, K)
                         float* __restrict__ num_ws,            // (D, F, K)
                         float* __restrict__ den_ws) {          // (D, K)
  const int wave   = blockIdx.x;
  const int kt     = wave % kKTiles;
  const int ft     = (wave / kKTiles) % kFTiles;
  const int bsplit = wave / (kKTiles * kFTiles);

  const int f0    = ft * 16;
  const int k0    = kt * 16;
  const int bbase = bsplit * kBChunk;

  const int lane = threadIdx.x;     // 0..31, wave32
  const int half = lane >> 4;       // 0: K=0,1   1: K=2,3
  const int l    = lane & 15;       // M index for A, N index for B
  const int kkA  = half << 1;       // 0 or 2

  v8f acc0 = {}, acc1 = {}, acc2 = {}, acc3 = {};
  float den0 = 0.f, den1 = 0.f, den2 = 0.f, den3 = 0.f;

  for (int c = 0; c < kBChunk; c += 4) {
    const int ba = bbase + c + kkA;     // this lane's first K-slot row of B
    const int bb = ba + 1;              // second K-slot row

    // Prefetch features for a future chunk (uniform condition; keeps EXEC full).
    if (c + 16 < kBChunk) {
      __builtin_prefetch(&features[(size_t)(bbase + c + 16 + kkA) * kF + f0 + l], 0, 3);
    }

    // A fragment: A[m,kk] = features[b, f0+m]
    v2f a;
    a.x = features[(size_t)ba * kF + f0 + l];
    a.y = features[(size_t)bb * kF + f0 + l];

    // Raw B fragment values: p[b, k0+n]
    const float p0 = probs[(size_t)ba * kK + k0 + l];
    const float p1 = probs[(size_t)bb * kK + k0 + l];
    const int   da = domains[ba];
    const int   db = domains[bb];

    v2f b;

    // domain 0
    b.x = (da == 0) ? p0 : 0.f;
    b.y = (db == 0) ? p1 : 0.f;
    den0 += b.x + b.y;
    acc0 = __builtin_amdgcn_wmma_f32_16x16x4_f32(false, a, false, b, (short)0,
                                                 acc0, false, false);
    // domain 1
    b.x = (da == 1) ? p0 : 0.f;
    b.y = (db == 1) ? p1 : 0.f;
    den1 += b.x + b.y;
    acc1 = __builtin_amdgcn_wmma_f32_16x16x4_f32(false, a, false, b, (short)0,
                                                 acc1, false, false);
    // domain 2
    b.x = (da == 2) ? p0 : 0.f;
    b.y = (db == 2) ? p1 : 0.f;
    den2 += b.x + b.y;
    acc2 = __builtin_amdgcn_wmma_f32_16x16x4_f32(false, a, false, b, (short)0,
                                                 acc2, false, false);
    // domain 3
    b.x = (da == 3) ? p0 : 0.f;
    b.y = (db == 3) ? p1 : 0.f;
    den3 += b.x + b.y;
    acc3 = __builtin_amdgcn_wmma_f32_16x16x4_f32(false, a, false, b, (short)0,
                                                 acc3, false, false);
  }

  // Flush accumulators: element r of lane -> (M = r + half*8, N = l)
#pragma unroll
  for (int r = 0; r < 8; ++r) {
    const int m = r + half * 8;
    const size_t base = (size_t)(f0 + m) * kK + (k0 + l);
    atomicAdd(&num_ws[0 * (size_t)kF * kK + base], acc0[r]);
    atomicAdd(&num_ws[1 * (size_t)kF * kK + base], acc1[r]);
    atomicAdd(&num_ws[2 * (size_t)kF * kK + base], acc2[r]);
    atomicAdd(&num_ws[3 * (size_t)kF * kK + base], acc3[r]);
  }

  // den does not depend on f: only f-tile 0 contributes (lane and lane+16 both
  // hold the same column n = l; atomics make the duplicate adds safe).
  if (ft == 0) {
    atomicAdd(&den_ws[0 * kK + k0 + l], den0);
    atomicAdd(&den_ws[1 * kK + k0 + l], den1);
    atomicAdd(&den_ws[2 * kK + k0 + l], den2);
    atomicAdd(&den_ws[3 * kK + k0 + l], den3);
  }
}

// ---------------------------------------------------------------------------
// Kernel 2: finalize. Global slice derived from the domain sums:
//   num_g = sum_d num_d ; den_g = sum_d den_d.
//   out[0]   = gstate*A + (num_g/(den_g+eps))*(1-A)
//   out[1+d] = dstate_d*A + (num_d/(den_d+eps))*(1-A)
// ---------------------------------------------------------------------------
__global__ __launch_bounds__(256)
void centroid_finalize(const float* __restrict__ gstate,     // (F, K)
                       const float* __restrict__ dstates,    // (D, F, K)
                       const float* __restrict__ num_ws,     // (D, F, K)
                       const float* __restrict__ den_ws,     // (D, K)
                       float* __restrict__ out) {            // (D+1, F, K)
  const int idx = blockIdx.x * blockDim.x + threadIdx.x;
  if (idx >= kF * kK) return;
  const int k = idx & (kK - 1);

  float num_g = 0.f, den_g = 0.f;
#pragma unroll
  for (int d = 0; d < kD; ++d) {
    const float n  = num_ws[(size_t)d * kF * kK + idx];
    const float dn = den_ws[d * kK + k];
    num_g += n;
    den_g += dn;
    const float cd = n / (dn + kEps);
    out[(size_t)(1 + d) * kF * kK + idx] =
        dstates[(size_t)d * kF * kK + idx] * kAlpha + cd * (1.f - kAlpha);
  }
  const float cg = num_g / (den_g + kEps);
  out[idx] = gstate[idx] * kAlpha + cg * (1.f - kAlpha);
}

extern "C" void kernel_launch(void* const* d_in, const int* in_sizes, int n_in,
                              void* d_out, int out_size, void* d_ws, size_t ws_size,
                              hipStream_t stream) {
  const float* features = (const float*)d_in[0];   // (B, F) f32
  const int*   domains  = (const int*)d_in[1];     // (1, B) int
  const float* probs    = (const float*)d_in[2];   // (B, K) f32
  const float* gstate   = (const float*)d_in[3];   // (F, K) f32
  const float* dstates  = (const float*)d_in[4];   // (D, F, K) f32
  float*       out      = (float*)d_out;           // (D+1, F, K) f32

  float* num_ws = (float*)d_ws;                          // D*F*K floats
  float* den_ws = num_ws + (size_t)kD * kF * kK;         // D*K floats

  const size_t zero_bytes =
      ((size_t)kD * kF * kK + (size_t)kD * kK) * sizeof(float);
  hipMemsetAsync(d_ws, 0, zero_bytes, stream);

  const int nwaves = kFTiles * kKTiles * kBSplit;  // 32*4*8 = 1024 waves
  centroid_accum_wmma<<<nwaves, 32, 0, stream>>>(features, domains, probs,
                                                 num_ws, den_ws);

  const int tot = kF * kK;
  centroid_finalize<<<(tot + 255) / 256, 256, 0, stream>>>(gstate, dstates,
                                                           num_ws, den_ws, out);
}